// GraphAttentionLayer_55903294325267
// MI455X (gfx1250) — compile-verified
//
#include <hip/hip_runtime.h>

#define BATCH 4
#define NNODE 16384
#define NEDGE 262144
#define FDIM  64

#define EXP_CLAMP   1000000.0f
#define DENOM_EPS   1e-10f
#define LRELU_ALPHA 0.01f

typedef float v2f __attribute__((ext_vector_type(2)));
typedef float v8f __attribute__((ext_vector_type(8)));

// ---------------------------------------------------------------------------
// Zero the output accumulator and the softmax denominator.
// ---------------------------------------------------------------------------
__global__ void gat_zero_kernel(float* __restrict__ out, float* __restrict__ denom) {
    int i = blockIdx.x * blockDim.x + threadIdx.x;
    if (i < BATCH * NNODE * FDIM) out[i] = 0.0f;
    if (i < BATCH * NNODE) denom[i] = 0.0f;
}

// ---------------------------------------------------------------------------
// Wh[b,n,o] = sum_f h[b,n,f] * W[o,f]   via V_WMMA_F32_16X16X4_F32.
// One wave per 16-row tile; the wave computes all 4 (16-wide) output column
// tiles, reusing its A fragment 4x per K-step. 16 K-steps * 4 tiles = 64 WMMAs.
//
// A (16x4, f32): lane L holds row M=L%16; VGPR v holds K = v + 2*(L/16).
// B (4x16, f32): lane L holds col  N=L%16; VGPR v holds K = v + 2*(L/16).
// D (16x16 f32): VGPR r: lanes 0-15 -> row r, lanes 16-31 -> row r+8.
// ---------------------------------------------------------------------------
__global__ __launch_bounds__(128)
void gat_gemm_wmma_kernel(const float* __restrict__ h,
                          const float* __restrict__ W,
                          float* __restrict__ Wh) {
    const int wave = threadIdx.x >> 5;
    const int lane = threadIdx.x & 31;
    const int tile = blockIdx.x * 4 + wave;          // over BATCH*NNODE/16 tiles
    const int tilesPerBatch = NNODE / 16;
    const int b     = tile / tilesPerBatch;
    const int nBase = (tile % tilesPerBatch) * 16;
    const int M    = lane & 15;
    const int half = lane >> 4;

    const float* __restrict__ hRow = h + ((size_t)b * NNODE + nBase + M) * FDIM;

    v8f acc0 = {}, acc1 = {}, acc2 = {}, acc3 = {};

#pragma unroll
    for (int k0 = 0; k0 < FDIM; k0 += 4) {
        // A fragment: this lane's row, K = k0 + 2*half + {0,1}
        v2f afrag;
        afrag.x = hRow[k0 + 2 * half + 0];
        afrag.y = hRow[k0 + 2 * half + 1];

        // B fragments: column o = oBase + M of W^T, i.e. W[o][k]
#pragma unroll
        for (int ot = 0; ot < 4; ++ot) {
            const float* __restrict__ wp = W + (ot * 16 + M) * FDIM + k0 + 2 * half;
            v2f bfrag;
            bfrag.x = wp[0];
            bfrag.y = wp[1];
            if (ot == 0)
                acc0 = __builtin_amdgcn_wmma_f32_16x16x4_f32(false, afrag, false, bfrag,
                                                             (short)0, acc0, false, false);
            else if (ot == 1)
                acc1 = __builtin_amdgcn_wmma_f32_16x16x4_f32(false, afrag, false, bfrag,
                                                             (short)0, acc1, false, false);
            else if (ot == 2)
                acc2 = __builtin_amdgcn_wmma_f32_16x16x4_f32(false, afrag, false, bfrag,
                                                             (short)0, acc2, false, false);
            else
                acc3 = __builtin_amdgcn_wmma_f32_16x16x4_f32(false, afrag, false, bfrag,
                                                             (short)0, acc3, false, false);
        }
    }

    // Store D: lane holds col = M, VGPR r holds row = r + 8*half.
#pragma unroll
    for (int r = 0; r < 8; ++r) {
        const int row = nBase + r + 8 * half;
        float* __restrict__ dst = Wh + ((size_t)b * NNODE + row) * FDIM + M;
        dst[0]  = acc0[r];
        dst[16] = acc1[r];
        dst[32] = acc2[r];
        dst[48] = acc3[r];
    }
}

// ---------------------------------------------------------------------------
// Per-node attention scores: s_c[n] = a[0:64].Wh[n], s_n[n] = a[64:128].Wh[n]
// ---------------------------------------------------------------------------
__global__ void gat_scores_kernel(const float* __restrict__ Wh,
                                  const float* __restrict__ avec,
                                  float* __restrict__ s_c,
                                  float* __restrict__ s_n) {
    int i = blockIdx.x * blockDim.x + threadIdx.x;
    if (i >= BATCH * NNODE) return;
    const float* __restrict__ row = Wh + (size_t)i * FDIM;
    float sc = 0.0f, sn = 0.0f;
#pragma unroll
    for (int o = 0; o < FDIM; ++o) {
        float v = row[o];
        sc += avec[o] * v;
        sn += avec[FDIM + o] * v;
    }
    s_c[i] = sc;
    s_n[i] = sn;
}

// ---------------------------------------------------------------------------
// Edge pass 1: att1 = leaky_relu(w*(s_c[ctr]+s_n[nbr])); x_exp = clip(exp);
// masked edges write 0. Accumulate denom via f32 atomics.
// ---------------------------------------------------------------------------
__global__ void gat_edge1_kernel(const int* __restrict__ edge,
                                 const int* __restrict__ edge_num,
                                 const float* __restrict__ ew,
                                 const float* __restrict__ s_c,
                                 const float* __restrict__ s_n,
                                 float* __restrict__ denom,
                                 float* __restrict__ xexp) {
    int t = blockIdx.x * blockDim.x + threadIdx.x;
    if (t >= BATCH * NEDGE) return;
    const int b = t / NEDGE;
    const int e = t - b * NEDGE;
    float xe = 0.0f;
    if (e < edge_num[b]) {
        const size_t eo = ((size_t)b * NEDGE + e) * 2;
        const int ctr = edge[eo + 0];
        const int nbr = edge[eo + 1];
        float att1 = ew[t] * (s_c[b * NNODE + ctr] + s_n[b * NNODE + nbr]);
        att1 = (att1 >= 0.0f) ? att1 : (LRELU_ALPHA * att1);
        xe = fminf(__expf(att1), EXP_CLAMP);
        atomicAdd(&denom[b * NNODE + ctr], xe);
    }
    xexp[t] = xe;
}

// ---------------------------------------------------------------------------
// Edge pass 2: out[ctr,:] += (x_exp/(eps+denom[ctr])) * Wh[nbr,:]
// 16 threads per edge, each owning a float4 slice of the 64-wide feature.
// ---------------------------------------------------------------------------
__global__ void gat_edge2_kernel(const int* __restrict__ edge,
                                 const float* __restrict__ Wh,
                                 const float* __restrict__ denom,
                                 const float* __restrict__ xexp,
                                 float* __restrict__ out) {
    int t = blockIdx.x * blockDim.x + threadIdx.x;   // over BATCH*NEDGE*16
    if (t >= BATCH * NEDGE * 16) return;
    const int f0  = (t & 15) * 4;
    const int eid = t >> 4;
    const float xe = xexp[eid];
    if (xe == 0.0f) return;                           // masked or zero weight
    const int b = eid / NEDGE;
    const int e = eid - b * NEDGE;
    const size_t eo = ((size_t)b * NEDGE + e) * 2;
    const int ctr = edge[eo + 0];
    const int nbr = edge[eo + 1];
    const float att = xe / (DENOM_EPS + denom[b * NNODE + ctr]);
    const float4 wn = *(const float4*)(Wh + ((size_t)b * NNODE + nbr) * FDIM + f0);
    float* __restrict__ dst = out + ((size_t)b * NNODE + ctr) * FDIM + f0;
    atomicAdd(dst + 0, att * wn.x);
    atomicAdd(dst + 1, att * wn.y);
    atomicAdd(dst + 2, att * wn.z);
    atomicAdd(dst + 3, att * wn.w);
}

// ---------------------------------------------------------------------------
// Final ReLU in place on the accumulated output.
// ---------------------------------------------------------------------------
__global__ void gat_relu_kernel(float* __restrict__ out) {
    int i = blockIdx.x * blockDim.x + threadIdx.x;
    if (i < BATCH * NNODE * FDIM) out[i] = fmaxf(out[i], 0.0f);
}

// ---------------------------------------------------------------------------
extern "C" void kernel_launch(void* const* d_in, const int* in_sizes, int n_in,
                              void* d_out, int out_size, void* d_ws, size_t ws_size,
                              hipStream_t stream) {
    const float* h    = (const float*)d_in[0];   // (B, N, F)
    const int*   edge = (const int*)  d_in[1];   // (B, E, 2)
    const int*   enum_= (const int*)  d_in[2];   // (B,)
    const float* ew   = (const float*)d_in[3];   // (B, E)
    const float* W    = (const float*)d_in[4];   // (F, F)
    const float* avec = (const float*)d_in[5];   // (1, 2F)
    float* out = (float*)d_out;                  // (B, N, F)

    // Workspace layout (floats)
    float* ws    = (float*)d_ws;
    float* Wh    = ws;                                   // B*N*F
    float* s_c   = Wh  + (size_t)BATCH * NNODE * FDIM;   // B*N
    float* s_n   = s_c + (size_t)BATCH * NNODE;          // B*N
    float* denom = s_n + (size_t)BATCH * NNODE;          // B*N
    float* xexp  = denom + (size_t)BATCH * NNODE;        // B*E

    // 1) zero out + denom
    {
        int n = BATCH * NNODE * FDIM;
        gat_zero_kernel<<<(n + 255) / 256, 256, 0, stream>>>(out, denom);
    }
    // 2) Wh = h @ W^T  (WMMA f32)
    {
        int tiles = (BATCH * NNODE) / 16;   // 4096 waves, 4 waves/block
        gat_gemm_wmma_kernel<<<tiles / 4, 128, 0, stream>>>(h, W, Wh);
    }
    // 3) per-node scores
    {
        int n = BATCH * NNODE;
        gat_scores_kernel<<<(n + 255) / 256, 256, 0, stream>>>(Wh, avec, s_c, s_n);
    }
    // 4) edge pass 1: logits -> exp -> denom atomics
    {
        int n = BATCH * NEDGE;
        gat_edge1_kernel<<<(n + 255) / 256, 256, 0, stream>>>(edge, enum_, ew, s_c, s_n,
                                                              denom, xexp);
    }
    // 5) edge pass 2: scatter-add att * Wh_n
    {
        long long n = (long long)BATCH * NEDGE * 16;
        gat_edge2_kernel<<<(int)((n + 255) / 256), 256, 0, stream>>>(edge, Wh, denom,
                                                                     xexp, out);
    }
    // 6) ReLU
    {
        int n = BATCH * NNODE * FDIM;
        gat_relu_kernel<<<(n + 255) / 256, 256, 0, stream>>>(out);
    }
}